// GlobalAttention_56745107915030
// MI455X (gfx1250) — compile-verified
//
#include <hip/hip_runtime.h>
#include <hip/hip_bf16.h>
#include <math.h>

#define B_ 16
#define T_ 64
#define S_ 256
#define D_ 512

typedef __attribute__((ext_vector_type(2))) float v2f;
typedef __attribute__((ext_vector_type(8))) float v8f;

static __device__ __forceinline__ float fast_tanh(float x) {
#if __has_builtin(__builtin_amdgcn_tanhf)
    return __builtin_amdgcn_tanhf(x);
#else
    return tanhf(x);
#endif
}

#define WMMA_F32(a, b, c) \
    __builtin_amdgcn_wmma_f32_16x16x4_f32(false, (a), false, (b), (short)0, (c), false, false)

// One wave accumulates a 16x64 fp32 tile of  C += A * W^T  via V_WMMA_F32_16X16X4_F32.
// - register-blocked 4x along N (one A-frag feeds 4 B-frags / 4 WMMAs per K-step)
// - software-pipelined: next K-step's 5 fragments are loaded before the current
//   step's WMMAs issue, so s_wait only covers loads one iteration back.
// ISA 32-bit A 16x4 layout: lanes 0-15 rows M, VGPR0=K(2*half), VGPR1=K(2*half+1);
// B (=W^T) mirrors it with lane = N. Both operands are row-major float2 loads at
// column (k + 2*half): a from A row (m0+lr), b_j from W row (n0+16j+lr).
// K is compile-time D_; LDW is the W row stride in floats (folds to immediates).
template <int LDW>
static __device__ __forceinline__ void wmma_rowrow4(const float* __restrict__ Arow,
                                                    const float* __restrict__ Wrow,
                                                    v8f acc[4]) {
    v2f a  = *(const v2f*)(Arow);
    v2f b0 = *(const v2f*)(Wrow + 0 * 16 * LDW);
    v2f b1 = *(const v2f*)(Wrow + 1 * 16 * LDW);
    v2f b2 = *(const v2f*)(Wrow + 2 * 16 * LDW);
    v2f b3 = *(const v2f*)(Wrow + 3 * 16 * LDW);
    for (int k = 4; k < D_; k += 4) {
        v2f an  = *(const v2f*)(Arow + k);
        v2f b0n = *(const v2f*)(Wrow + 0 * 16 * LDW + k);
        v2f b1n = *(const v2f*)(Wrow + 1 * 16 * LDW + k);
        v2f b2n = *(const v2f*)(Wrow + 2 * 16 * LDW + k);
        v2f b3n = *(const v2f*)(Wrow + 3 * 16 * LDW + k);
        acc[0] = WMMA_F32(a, b0, acc[0]);
        acc[1] = WMMA_F32(a, b1, acc[1]);
        acc[2] = WMMA_F32(a, b2, acc[2]);
        acc[3] = WMMA_F32(a, b3, acc[3]);
        a = an; b0 = b0n; b1 = b1n; b2 = b2n; b3 = b3n;
    }
    acc[0] = WMMA_F32(a, b0, acc[0]);
    acc[1] = WMMA_F32(a, b1, acc[1]);
    acc[2] = WMMA_F32(a, b2, acc[2]);
    acc[3] = WMMA_F32(a, b3, acc[3]);
}

// C[M,512] = A[M,512] * W[512,512]^T + bias (bias may be null). 16x64 tile/wave.
__global__ void k_gemm_bias(const float* __restrict__ A, const float* __restrict__ W,
                            const float* __restrict__ bias, float* __restrict__ C,
                            int M) {
    const int N = D_, K = D_;
    int wave = (blockIdx.x * blockDim.x + threadIdx.x) >> 5;
    int lane = threadIdx.x & 31;
    const int ntg = N >> 6;                       // groups of 64 columns
    if (wave >= (M >> 4) * ntg) return;           // wave-uniform: EXEC stays all-1s
    int m0 = (wave / ntg) << 4, n0 = (wave % ntg) << 6;
    int half = lane >> 4, lr = lane & 15;
    v8f acc[4] = {v8f{}, v8f{}, v8f{}, v8f{}};
    wmma_rowrow4<D_>(A + (m0 + lr) * K + 2 * half,
                     W + (n0 + lr) * K + 2 * half, acc);
#pragma unroll
    for (int j = 0; j < 4; ++j) {
        int n = n0 + j * 16 + lr;
        float bv = bias ? bias[n] : 0.0f;
#pragma unroll
        for (int r = 0; r < 8; ++r) {
            int m = m0 + r + 8 * half;            // ISA C/D layout: VGPR r -> M = r + 8*half
            C[m * N + n] = acc[j][r] + bv;
        }
    }
}

// UHt[b][e][s] = sum_d context[b][s][d] * Wc[e][d]   (transposed store, no bias)
__global__ void k_gemm_uht(const float* __restrict__ ctx, const float* __restrict__ Wc,
                           float* __restrict__ UHt) {
    const int K = D_;
    int wave = (blockIdx.x * blockDim.x + threadIdx.x) >> 5;  // 2048 waves exactly
    int lane = threadIdx.x & 31;
    const int ntg = D_ >> 6;                                  // 8 column groups
    int m0 = (wave / ntg) << 4, n0 = (wave % ntg) << 6;
    int half = lane >> 4, lr = lane & 15;
    v8f acc[4] = {v8f{}, v8f{}, v8f{}, v8f{}};
    wmma_rowrow4<D_>(ctx + (m0 + lr) * K + 2 * half,
                     Wc + (n0 + lr) * K + 2 * half, acc);
#pragma unroll
    for (int j = 0; j < 4; ++j) {
        int n = n0 + j * 16 + lr;
#pragma unroll
        for (int r = 0; r < 8; ++r) {
            int m = m0 + r + 8 * half;            // m = b*S + s (tile never straddles b)
            int b = m >> 8, s = m & (S_ - 1);
            UHt[b * (D_ * S_) + n * S_ + s] = acc[j][r];
        }
    }
}

// attn_h = Cc * Wo[:, :D]^T + input * Wo[:, D:]^T + bo   (Wo row stride = 2D)
__global__ void k_gemm_out(const float* __restrict__ Cc, const float* __restrict__ inp,
                           const float* __restrict__ Wo, const float* __restrict__ bo,
                           float* __restrict__ out) {
    const int N = D_;
    int wave = (blockIdx.x * blockDim.x + threadIdx.x) >> 5;  // 512 waves exactly
    int lane = threadIdx.x & 31;
    const int ntg = N >> 6;
    int m0 = (wave / ntg) << 4, n0 = (wave % ntg) << 6;
    int half = lane >> 4, lr = lane & 15;
    const float* wrow = Wo + (n0 + lr) * (2 * D_) + 2 * half;
    v8f acc[4] = {v8f{}, v8f{}, v8f{}, v8f{}};
    wmma_rowrow4<2 * D_>(Cc  + (m0 + lr) * D_ + 2 * half, wrow,      acc);
    wmma_rowrow4<2 * D_>(inp + (m0 + lr) * D_ + 2 * half, wrow + D_, acc);
#pragma unroll
    for (int j = 0; j < 4; ++j) {
        int n = n0 + j * 16 + lr;
        float bv = bo[n];
#pragma unroll
        for (int r = 0; r < 8; ++r) {
            int m = m0 + r + 8 * half;
            out[m * N + n] = acc[j][r] + bv;
        }
    }
}

// One block per (b,t) row: scores -> masked softmax -> context vector c.
__global__ void k_attend(const float* __restrict__ WQ, const float* __restrict__ UHt,
                         const float* __restrict__ ctx, const float* __restrict__ vvec,
                         const int* __restrict__ lens,
                         float* __restrict__ alignOut, float* __restrict__ Cc) {
    __shared__ float s_wq[D_];
    __shared__ float s_v[D_];
    __shared__ float s_red[S_];
    __shared__ float s_a[S_];

    const int row = blockIdx.x;          // b*T + t
    const int b   = row / T_;
    const int tid = threadIdx.x;         // 256 threads, thread = s

    for (int i = tid; i < D_; i += S_) { s_wq[i] = WQ[row * D_ + i]; s_v[i] = vvec[i]; }
    __syncthreads();

    const int s = tid;
    const float* uh = UHt + b * (D_ * S_) + s;   // stride S_ over d -> coalesced lanes
    float a0 = 0.f, a1 = 0.f, a2 = 0.f, a3 = 0.f;
    for (int d = 0; d < D_; d += 4) {
        a0 += s_v[d + 0] * fast_tanh(s_wq[d + 0] + uh[(d + 0) * S_]);
        a1 += s_v[d + 1] * fast_tanh(s_wq[d + 1] + uh[(d + 1) * S_]);
        a2 += s_v[d + 2] * fast_tanh(s_wq[d + 2] + uh[(d + 2) * S_]);
        a3 += s_v[d + 3] * fast_tanh(s_wq[d + 3] + uh[(d + 3) * S_]);
    }
    float score = (a0 + a1) + (a2 + a3);

    const int len = lens[b];
    const bool valid = (s < len);

    s_red[tid] = valid ? score : -INFINITY;
    __syncthreads();
    for (int off = 128; off > 0; off >>= 1) {
        if (tid < off) s_red[tid] = fmaxf(s_red[tid], s_red[tid + off]);
        __syncthreads();
    }
    float mx = s_red[0];
    __syncthreads();

    float e = valid ? __expf(score - mx) : 0.0f;
    s_red[tid] = e;
    __syncthreads();
    for (int off = 128; off > 0; off >>= 1) {
        if (tid < off) s_red[tid] += s_red[tid + off];
        __syncthreads();
    }
    float a = e / s_red[0];

    alignOut[row * S_ + s] = a;
    s_a[s] = a;
    __syncthreads();

    const float* cb = ctx + b * (S_ * D_);
    for (int d = tid; d < D_; d += S_) {
        float sum = 0.0f;
#pragma unroll 8
        for (int ss = 0; ss < S_; ++ss) sum += s_a[ss] * cb[ss * D_ + d];
        Cc[row * D_ + d] = sum;
    }
}

extern "C" void kernel_launch(void* const* d_in, const int* in_sizes, int n_in,
                              void* d_out, int out_size, void* d_ws, size_t ws_size,
                              hipStream_t stream) {
    const float* input   = (const float*)d_in[0];
    const float* context = (const float*)d_in[1];
    const int*   lens    = (const int*)d_in[2];
    // d_in[3] = context_max_len (compile-time S_)
    const float* Wq = (const float*)d_in[4];
    const float* bq = (const float*)d_in[5];
    const float* Wc = (const float*)d_in[6];
    const float* v  = (const float*)d_in[7];
    const float* Wo = (const float*)d_in[8];
    const float* bo = (const float*)d_in[9];

    float* out_attn  = (float*)d_out;                 // [B,T,D]
    float* out_align = out_attn + B_ * T_ * D_;       // [B,T,S]

    float* ws  = (float*)d_ws;
    float* WQ  = ws;                                  // [B*T, D]   2 MB
    float* UHt = WQ + B_ * T_ * D_;                   // [B, D, S]  8 MB
    float* Cc  = UHt + B_ * D_ * S_;                  // [B*T, D]   2 MB

    // 1) WQ = input*Wq^T + bq : 64 m-tiles x 8 n-groups = 512 waves -> 64 blocks
    k_gemm_bias<<<64, 256, 0, stream>>>(input, Wq, bq, WQ, B_ * T_);
    // 2) UHt = (context*Wc^T) transposed : 256 x 8 = 2048 waves -> 256 blocks
    k_gemm_uht<<<256, 256, 0, stream>>>(context, Wc, UHt);
    // 3) scores + softmax + context vector, one block per (b,t)
    k_attend<<<B_ * T_, S_, 0, stream>>>(WQ, UHt, context, v, lens, out_align, Cc);
    // 4) attn_h = [Cc, input]*Wo^T + bo : 64 x 8 = 512 waves -> 64 blocks
    k_gemm_out<<<64, 256, 0, stream>>>(Cc, input, Wo, bo, out_attn);
}